// SVDHead_66022237274596
// MI455X (gfx1250) — compile-verified
//
#include <hip/hip_runtime.h>
#include <math.h>

typedef __attribute__((ext_vector_type(2))) float v2f;
typedef __attribute__((ext_vector_type(8))) float v8f;
typedef __attribute__((ext_vector_type(4))) int   v4i;

constexpr int kB = 8;
constexpr int kC = 512;
constexpr int kN = 2048;
constexpr int ASTR = kC + 2;   // 514: pad so b64 LDS reads across 16 lanes hit distinct bank pairs
constexpr float kInvSqrtC = 0.04419417382415922f; // 1/sqrt(512)

#if defined(__has_builtin)
#if __has_builtin(__builtin_amdgcn_global_load_async_to_lds_b128)
#define USE_ASYNC_LDS 1
#endif
#endif

#if USE_ASYNC_LDS
__device__ __forceinline__ void async_copy_b128(const float* g, float* l) {
  // builtin signature (from hipcc diagnostic): (v4i AS1*, v4i AS3*, imm int, imm int)
  v4i* gg = (v4i*)(void*)g;   // strip const, retype (still generic)
  v4i* ll = (v4i*)(void*)l;
  __builtin_amdgcn_global_load_async_to_lds_b128(
      (__attribute__((address_space(1))) v4i*)gg,
      (__attribute__((address_space(3))) v4i*)ll, 0, 0);
}
#endif

// ---------------------------------------------------------------------------
// Kernel 1: fused scores-GEMM (fp32 WMMA) + gumbel noise + row argmax.
// Block = 128 threads (4 waves), owns 32 score rows (2 WMMA sub-tiles) of one
// batch; waves split the 128 column tiles 4-ways. B panels are staged with
// gfx1250 async global->LDS b128 copies (ASYNCcnt) in natural [c][m] layout;
// scores are never written to memory.
// ---------------------------------------------------------------------------
__global__ __launch_bounds__(128) void scores_argmax_kernel(
    const float* __restrict__ src_emb, const float* __restrict__ tgt_emb,
    const float* __restrict__ gumbel, int* __restrict__ idx_out) {
  __shared__ __align__(16) float As[32 * ASTR];       // A panel transposed: As[row][c]
  __shared__ __align__(16) float Bs[4 * 128 * 16];    // per-wave B chunk: [c_local][m]
  __shared__ float redVal[4][32];
  __shared__ int   redIdx[4][32];

  const int tile = blockIdx.x & 63;      // N/32 = 64 row-tiles
  const int b    = blockIdx.x >> 6;
  const int n0   = tile << 5;

  // Stage A panel: As[row][c] = src_emb[b][c][n0+row]
  {
    const int row = threadIdx.x & 31;
    const int cb  = threadIdx.x >> 5;    // 0..3
    const float* srcb = src_emb + (size_t)b * kC * kN + (size_t)(n0 + row);
    for (int c = cb; c < kC; c += 4)
      As[row * ASTR + c] = srcb[(size_t)c * kN];
  }
  __syncthreads();

  const int wave = threadIdx.x >> 5;
  const int lane = threadIdx.x & 31;
  const int half = lane >> 4;            // K-half for WMMA A/B operands
  const int col  = lane & 15;            // M index for A, N index for B/C/D

  float* bsn = &Bs[wave * 128 * 16];
  const float* tgtb = tgt_emb + (size_t)b * kC * kN;

  float bestv[2][8];
  int   bestm[2][8];
#pragma unroll
  for (int s = 0; s < 2; ++s)
#pragma unroll
    for (int j = 0; j < 8; ++j) { bestv[s][j] = -INFINITY; bestm[s][j] = 0; }

  // staging geometry: 128 rows x 64B per chunk; 4 lanes per row, 16B per lane
  const int srow = lane >> 2;            // + 8*t
  const int smo  = (lane & 3) << 2;

  for (int mt = wave; mt < kN / 16; mt += 4) {
    const int m0 = mt << 4;
    v8f acc0 = {0.f, 0.f, 0.f, 0.f, 0.f, 0.f, 0.f, 0.f};
    v8f acc1 = {0.f, 0.f, 0.f, 0.f, 0.f, 0.f, 0.f, 0.f};

    for (int cc = 0; cc < kC; cc += 128) {
#if USE_ASYNC_LDS
      // ASYNC stores to LDS are NOT ordered with DS reads: drain reads of the
      // previous chunk before overwriting the buffer.
      asm volatile("s_wait_dscnt 0x0" ::: "memory");
#endif
      // Stage B chunk in natural layout: bsn[c][m] = tgt_emb[b][cc+c][m0+m]
#pragma unroll 4
      for (int t = 0; t < 16; ++t) {
        const int r = (t << 3) + srow;
        const float* g = tgtb + (size_t)(cc + r) * kN + (size_t)(m0 + smo);
        float* l = &bsn[r * 16 + smo];
#if USE_ASYNC_LDS
        async_copy_b128(g, l);
#else
        *(float4*)l = *(const float4*)g;
#endif
      }
#if USE_ASYNC_LDS
      asm volatile("s_wait_asynccnt 0x0" ::: "memory");
#endif
      __builtin_amdgcn_wave_barrier();

      // ISA 32-bit 16x4 A layout: lane(half,M=col): a = A[M][2*half + {0,1}]
      // ISA 4x16 B layout: b = B[2*half + {0,1}][col] -> ds_load_2addr_b32 pair
#pragma unroll
      for (int k = 0; k < 128; k += 4) {
        const float* bp = &bsn[(k + 2 * half) * 16 + col];
        v2f bv = {bp[0], bp[16]};
        v2f a0 = *(const v2f*)&As[col * ASTR + cc + k + 2 * half];
        v2f a1 = *(const v2f*)&As[(16 + col) * ASTR + cc + k + 2 * half];
        acc0 = __builtin_amdgcn_wmma_f32_16x16x4_f32(false, a0, false, bv,
                                                     (short)0, acc0, false, false);
        acc1 = __builtin_amdgcn_wmma_f32_16x16x4_f32(false, a1, false, bv,
                                                     (short)0, acc1, false, false);
      }
      __builtin_amdgcn_wave_barrier();
    }

    // C/D layout: vgpr j, lanes 0-15 -> M=j, lanes 16-31 -> M=8+j; N=col.
#pragma unroll
    for (int s = 0; s < 2; ++s) {
#pragma unroll
      for (int j = 0; j < 8; ++j) {
        const int n = n0 + s * 16 + j + 8 * half;
        float u = gumbel[(size_t)(b * kN + n) * kN + (size_t)(m0 + col)];
        u = fminf(fmaxf(u, 1e-6f), 1.0f - 1e-6f);
        const float g = -logf(-logf(u));
        const float a = s ? acc1[j] : acc0[j];
        const float v = a * kInvSqrtC + g;
        if (v > bestv[s][j]) { bestv[s][j] = v; bestm[s][j] = m0 + col; }
      }
    }
  }

  // argmax across the 16 lanes sharing each row (xor masks stay inside groups)
#pragma unroll
  for (int s = 0; s < 2; ++s) {
#pragma unroll
    for (int j = 0; j < 8; ++j) {
      float v = bestv[s][j];
      int   m = bestm[s][j];
      for (int off = 8; off >= 1; off >>= 1) {
        float ov = __shfl_xor(v, off, 32);
        int   om = __shfl_xor(m, off, 32);
        if (ov > v || (ov == v && om < m)) { v = ov; m = om; }
      }
      if (col == 0) {
        redVal[wave][s * 16 + j + 8 * half] = v;
        redIdx[wave][s * 16 + j + 8 * half] = m;
      }
    }
  }
  __syncthreads();
  if (threadIdx.x < 32) {
    const int r = threadIdx.x;
    float v = redVal[0][r];
    int   m = redIdx[0][r];
    for (int w = 1; w < 4; ++w) {
      const float ov = redVal[w][r];
      const int   om = redIdx[w][r];
      if (ov > v || (ov == v && om < m)) { v = ov; m = om; }
    }
    idx_out[b * kN + n0 + r] = m;
  }
}

// ---------------------------------------------------------------------------
// Kernel 2: per-batch means + 3x3 covariance H from the gathered
// correspondences (deterministic shuffle/LDS tree reduction, no float atomics).
// stats[b] = { H[9], mu_src[3], mu_corr[3], pad }
// ---------------------------------------------------------------------------
__global__ __launch_bounds__(256) void cov_kernel(
    const float* __restrict__ src, const float* __restrict__ tgt,
    const int* __restrict__ idx, float* __restrict__ stats) {
  const int b = blockIdx.x;
  const float* S = src + (size_t)b * 3 * kN;
  const float* T = tgt + (size_t)b * 3 * kN;
  const int*   I = idx + b * kN;

  float a[15];
#pragma unroll
  for (int k = 0; k < 15; ++k) a[k] = 0.f;

  for (int n = threadIdx.x; n < kN; n += blockDim.x) {
    const float s0 = S[n], s1 = S[kN + n], s2 = S[2 * kN + n];
    const int m = I[n];
    const float c0 = T[m], c1 = T[kN + m], c2 = T[2 * kN + m];
    a[0] += s0; a[1] += s1; a[2] += s2;
    a[3] += c0; a[4] += c1; a[5] += c2;
    a[6]  += s0 * c0; a[7]  += s0 * c1; a[8]  += s0 * c2;
    a[9]  += s1 * c0; a[10] += s1 * c1; a[11] += s1 * c2;
    a[12] += s2 * c0; a[13] += s2 * c1; a[14] += s2 * c2;
  }

  __shared__ float wred[8][15];
  const int wave = threadIdx.x >> 5, lane = threadIdx.x & 31;
#pragma unroll
  for (int k = 0; k < 15; ++k) {
    float v = a[k];
    for (int off = 16; off >= 1; off >>= 1) v += __shfl_xor(v, off, 32);
    if (lane == 0) wred[wave][k] = v;
  }
  __syncthreads();
  if (threadIdx.x == 0) {
    float tot[15];
    for (int k = 0; k < 15; ++k) {
      float v = 0.f;
      for (int w = 0; w < 8; ++w) v += wred[w][k];
      tot[k] = v;
    }
    const float invN = 1.0f / (float)kN;
    const float mus[3] = {tot[0] * invN, tot[1] * invN, tot[2] * invN};
    const float muc[3] = {tot[3] * invN, tot[4] * invN, tot[5] * invN};
    float* o = stats + b * 16;
    for (int c = 0; c < 3; ++c)
      for (int d = 0; d < 3; ++d)
        o[c * 3 + d] = tot[6 + c * 3 + d] - (float)kN * mus[c] * muc[d];
    o[9]  = mus[0]; o[10] = mus[1]; o[11] = mus[2];
    o[12] = muc[0]; o[13] = muc[1]; o[14] = muc[2];
    o[15] = 0.f;
  }
}

// ---------------------------------------------------------------------------
// Kernel 3: per-batch 3x3 SVD (Jacobi on H^T H) + Kabsch R, t. One lane per batch.
// ---------------------------------------------------------------------------
__device__ inline float det3(const float M[3][3]) {
  return M[0][0] * (M[1][1] * M[2][2] - M[1][2] * M[2][1])
       - M[0][1] * (M[1][0] * M[2][2] - M[1][2] * M[2][0])
       + M[0][2] * (M[1][0] * M[2][1] - M[1][1] * M[2][0]);
}

__global__ void svd_head_kernel(const float* __restrict__ stats, float* __restrict__ out) {
  const int b = threadIdx.x;
  if (b >= kB) return;
  const float* s = stats + b * 16;
  float H[3][3], mus[3], muc[3];
  for (int i = 0; i < 3; ++i) {
    for (int j = 0; j < 3; ++j) H[i][j] = s[i * 3 + j];
    mus[i] = s[9 + i];
    muc[i] = s[12 + i];
  }

  // A = H^T H (symmetric, eigvecs = right singular vectors V)
  float A[3][3];
  for (int i = 0; i < 3; ++i)
    for (int j = 0; j < 3; ++j)
      A[i][j] = H[0][i] * H[0][j] + H[1][i] * H[1][j] + H[2][i] * H[2][j];

  float V[3][3] = {{1.f, 0.f, 0.f}, {0.f, 1.f, 0.f}, {0.f, 0.f, 1.f}};
  const int P[3] = {0, 0, 1}, Q[3] = {1, 2, 2};
  for (int sweep = 0; sweep < 8; ++sweep) {
    for (int r = 0; r < 3; ++r) {
      const int p = P[r], q = Q[r];
      const float apq = A[p][q];
      if (fabsf(apq) < 1e-20f) continue;
      const float tau = (A[q][q] - A[p][p]) / (2.f * apq);
      const float t   = (tau >= 0.f ? 1.f : -1.f) / (fabsf(tau) + sqrtf(1.f + tau * tau));
      const float c   = 1.f / sqrtf(1.f + t * t);
      const float sn  = t * c;
      for (int k = 0; k < 3; ++k) {       // A <- A G
        const float akp = A[k][p], akq = A[k][q];
        A[k][p] = c * akp - sn * akq;
        A[k][q] = sn * akp + c * akq;
      }
      for (int k = 0; k < 3; ++k) {       // A <- G^T A
        const float apk = A[p][k], aqk = A[q][k];
        A[p][k] = c * apk - sn * aqk;
        A[q][k] = sn * apk + c * aqk;
      }
      for (int k = 0; k < 3; ++k) {       // V <- V G
        const float vkp = V[k][p], vkq = V[k][q];
        V[k][p] = c * vkp - sn * vkq;
        V[k][q] = sn * vkp + c * vkq;
      }
    }
  }

  // sort eigenvalues descending (match svd's descending singular values)
  float lam[3] = {A[0][0], A[1][1], A[2][2]};
  for (int i = 0; i < 2; ++i)
    for (int j = i + 1; j < 3; ++j)
      if (lam[j] > lam[i]) {
        const float tl = lam[i]; lam[i] = lam[j]; lam[j] = tl;
        for (int k = 0; k < 3; ++k) {
          const float tv = V[k][i]; V[k][i] = V[k][j]; V[k][j] = tv;
        }
      }

  // U columns: u_i = H v_i / s_i
  float U[3][3];
  for (int i = 0; i < 3; ++i) {
    float u0 = H[0][0] * V[0][i] + H[0][1] * V[1][i] + H[0][2] * V[2][i];
    float u1 = H[1][0] * V[0][i] + H[1][1] * V[1][i] + H[1][2] * V[2][i];
    float u2 = H[2][0] * V[0][i] + H[2][1] * V[1][i] + H[2][2] * V[2][i];
    const float nrm = sqrtf(u0 * u0 + u1 * u1 + u2 * u2);
    if (nrm > 1e-12f) {
      const float r = 1.f / nrm;
      U[0][i] = u0 * r; U[1][i] = u1 * r; U[2][i] = u2 * r;
    } else if (i == 2) {                  // rank-deficient fallback: u2 = u0 x u1
      U[0][2] = U[1][0] * U[2][1] - U[2][0] * U[1][1];
      U[1][2] = U[2][0] * U[0][1] - U[0][0] * U[2][1];
      U[2][2] = U[0][0] * U[1][1] - U[1][0] * U[0][1];
    } else {
      U[0][i] = (i == 0); U[1][i] = (i == 1); U[2][i] = 0.f;
    }
  }

  // R = V diag(1,1,det(V U^T)) U^T ;  t = -R mu_src + mu_corr
  float R0[3][3];
  for (int i = 0; i < 3; ++i)
    for (int j = 0; j < 3; ++j)
      R0[i][j] = V[i][0] * U[j][0] + V[i][1] * U[j][1] + V[i][2] * U[j][2];
  const float d = det3(R0);
  float R[3][3];
  for (int i = 0; i < 3; ++i)
    for (int j = 0; j < 3; ++j)
      R[i][j] = V[i][0] * U[j][0] + V[i][1] * U[j][1] + d * V[i][2] * U[j][2];

  for (int i = 0; i < 3; ++i)
    for (int j = 0; j < 3; ++j)
      out[b * 9 + i * 3 + j] = R[i][j];
  for (int i = 0; i < 3; ++i)
    out[kB * 9 + b * 3 + i] =
        muc[i] - (R[i][0] * mus[0] + R[i][1] * mus[1] + R[i][2] * mus[2]);
}

// ---------------------------------------------------------------------------
extern "C" void kernel_launch(void* const* d_in, const int* in_sizes, int n_in,
                              void* d_out, int out_size, void* d_ws, size_t ws_size,
                              hipStream_t stream) {
  const float* src_emb = (const float*)d_in[0];  // [8,512,2048]
  const float* tgt_emb = (const float*)d_in[1];  // [8,512,2048]
  const float* src     = (const float*)d_in[2];  // [8,3,2048]
  const float* tgt     = (const float*)d_in[3];  // [8,3,2048]
  // d_in[4] = temperature (positive scalar per batch; argmax-invariant)
  const float* gumbel  = (const float*)d_in[5];  // [16384,2048]

  int*   idx   = (int*)d_ws;                                    // 64 KB
  float* stats = (float*)((char*)d_ws + (size_t)kB * kN * 4);   // 512 B

  scores_argmax_kernel<<<dim3(kB * (kN / 32)), dim3(128), 0, stream>>>(
      src_emb, tgt_emb, gumbel, idx);
  cov_kernel<<<dim3(kB), dim3(256), 0, stream>>>(src, tgt, idx, stats);
  svd_head_kernel<<<dim3(1), dim3(32), 0, stream>>>(stats, (float*)d_out);
}